// TopKFreq_4449586118737
// MI455X (gfx1250) — compile-verified
//
#include <hip/hip_runtime.h>
#include <hip/hip_bf16.h>
#include <math.h>

typedef __attribute__((ext_vector_type(2))) float v2f;
typedef __attribute__((ext_vector_type(4))) float v4f;
typedef __attribute__((ext_vector_type(8))) float v8f;

#define KCOEF 5
#define LLEN  1024
#define PI_F  3.14159265358979323846f

// ---------------------------------------------------------------------------
// Kernel 0: w[l] = sum_k C[k,l] * fc_w[k]   (collapse DCT rows + conv weight)
// C[0,l] = sqrt(1/L); C[k,l] = cos(pi*(l+0.5)*k/L)*sqrt(2/L) for k>=1.
// ---------------------------------------------------------------------------
__global__ void topk_prep_w(const float* __restrict__ fc_w,
                            float* __restrict__ w, int L) {
    int l = blockIdx.x * blockDim.x + threadIdx.x;
    if (l >= L) return;
    float invL  = 1.0f / (float)L;
    float s     = fc_w[0] * sqrtf(invL);
    float base  = (l + 0.5f) * PI_F * invL;
    float scale = sqrtf(2.0f * invL);
#pragma unroll
    for (int k = 1; k < KCOEF; ++k)
        s += fc_w[k] * (cosf(base * (float)k) * scale);
    w[l] = s;
}

// ---------------------------------------------------------------------------
// Kernel 1: att[r] = x[r,:] . w[:] + fc_b   via V_WMMA_F32_16X16X4_F32 with
// double-buffered GLOBAL_LOAD_ASYNC_TO_LDS_B128 staging (ASYNCcnt-tracked).
// One wave per 16-row tile; 256-thread block = 8 waves = 128 rows.
// Per 32-column chunk: 4 async b128 ops (coalesced: 8 lanes cover one 128B
// row) land the 16x32 f32 tile in LDS while the previous chunk feeds 8
// rank-4 f32 WMMAs. Async loads complete in order, so with 8 ops in flight
// "s_wait_asynccnt 4" guarantees the current chunk's 4 ops have landed.
// B broadcasts w across all 16 N columns, so every output column = the dot.
// ---------------------------------------------------------------------------
__global__ void topk_gemv_wmma(const float* __restrict__ x,
                               const float* __restrict__ w,
                               const float* __restrict__ fc_b,
                               float* __restrict__ att, int L) {
    __shared__ float tile[8][2][16 * 32];     // 2 KB x 2 buffers per wave = 32 KB
    const int lane = threadIdx.x & 31;
    const int wave = threadIdx.x >> 5;
    const int r0   = (blockIdx.x * 8 + wave) * 16;

    const int m    = lane & 15;               // A-matrix row this lane owns
    const int koff = (lane >> 4) * 2;         // lane<16 -> K{0,1}; lane>=16 -> K{2,3}
    const int rowb = lane >> 3;               // 0..3: row-within-group for loads
    const int colb = (lane & 7) * 4;          // 0..28: 8 lanes span one row

    // LDS aperture maps generic addr[31:0] directly to the DS address space.
    const unsigned ldsbase = (unsigned)(uintptr_t)&tile[wave][0][0];

    const int NCHUNK = L / 32;                // 32 chunks of 16x32 f32

    // ---- prologue: async-stage chunk 0 into buffer 0 ----
#pragma unroll
    for (int i = 0; i < 4; ++i) {
        int row = i * 4 + rowb;
        unsigned long long g =
            (unsigned long long)(uintptr_t)(x + (size_t)(r0 + row) * L + colb);
        unsigned la = ldsbase + (unsigned)((row * 32 + colb) * 4);
        asm volatile("global_load_async_to_lds_b128 %0, %1, off"
                     :: "v"(la), "v"(g) : "memory");
    }

    v8f acc = {};                             // 16x16 f32 accumulator (8 VGPRs)

    for (int chunk = 0; chunk < NCHUNK; ++chunk) {
        const int cur = chunk & 1;
        if (chunk + 1 < NCHUNK) {
            // issue next chunk into the other buffer (8 async ops in flight)
            int c1 = (chunk + 1) * 32;
#pragma unroll
            for (int i = 0; i < 4; ++i) {
                int row = i * 4 + rowb;
                unsigned long long g = (unsigned long long)(uintptr_t)
                    (x + (size_t)(r0 + row) * L + c1 + colb);
                unsigned la = ldsbase +
                    (unsigned)(((cur ^ 1) * 512 + row * 32 + colb) * 4);
                asm volatile("global_load_async_to_lds_b128 %0, %1, off"
                             :: "v"(la), "v"(g) : "memory");
            }
            asm volatile("s_wait_asynccnt 0x4" ::: "memory");
        } else {
            asm volatile("s_wait_asynccnt 0x0" ::: "memory");
        }

        const float* ldsr = &tile[wave][cur][0];
        const int c0 = chunk * 32;
        // ---- 8 rank-4 WMMA updates over this chunk ----
#pragma unroll
        for (int s = 0; s < 8; ++s) {
            int k0 = s * 4 + koff;
            v2f a = *(const v2f*)(ldsr + m * 32 + k0);   // ds_load_b64
            v2f b = *(const v2f*)(w + c0 + k0);          // w broadcast over N
            acc = __builtin_amdgcn_wmma_f32_16x16x4_f32(
                /*neg_a=*/false, a, /*neg_b=*/false, b,
                /*c_mod=*/(short)0, acc, /*reuse_a=*/false, /*reuse_b=*/false);
        }
    }

    // D layout: VGPR v holds M=v (lanes 0-15) / M=v+8 (lanes 16-31); every
    // column N is identical, so lanes 0 and 16 write the 16 row results.
    float bias = fc_b[0];
    if ((lane & 15) == 0) {
        int base = r0 + ((lane >> 4) ? 8 : 0);
#pragma unroll
        for (int v = 0; v < 8; ++v)
            att[base + v] = acc[v] + bias;
    }
}

// ---------------------------------------------------------------------------
// Kernel 2: deterministic per-block partial sums of att and att^2.
// 64 blocks x 256 threads, contiguous 512-element slab per block.
// ---------------------------------------------------------------------------
__global__ void topk_reduce_partial(const float* __restrict__ att,
                                    float* __restrict__ psum,
                                    float* __restrict__ psumsq, int n) {
    __shared__ float ss[256];
    __shared__ float sq[256];
    int t = threadIdx.x;
    int base = blockIdx.x * 512;
    float s = 0.f, q = 0.f;
#pragma unroll
    for (int i = 0; i < 2; ++i) {
        int idx = base + i * 256 + t;
        if (idx < n) {
            float a = att[idx];
            s += a;
            q += a * a;
        }
    }
    ss[t] = s; sq[t] = q;
    __syncthreads();
    for (int stride = 128; stride > 0; stride >>= 1) {
        if (t < stride) { ss[t] += ss[t + stride]; sq[t] += sq[t + stride]; }
        __syncthreads();
    }
    if (t == 0) { psum[blockIdx.x] = ss[0]; psumsq[blockIdx.x] = sq[0]; }
}

// ---------------------------------------------------------------------------
// Kernel 3: final reduce of 64 partials -> stats[0]=mean, stats[1]=rsigma.
// ---------------------------------------------------------------------------
__global__ void topk_reduce_final(const float* __restrict__ psum,
                                  const float* __restrict__ psumsq,
                                  float* __restrict__ stats, int n) {
    __shared__ float ss[64];
    __shared__ float sq[64];
    int t = threadIdx.x;
    ss[t] = psum[t]; sq[t] = psumsq[t];
    __syncthreads();
    for (int stride = 32; stride > 0; stride >>= 1) {
        if (t < stride) { ss[t] += ss[t + stride]; sq[t] += sq[t + stride]; }
        __syncthreads();
    }
    if (t == 0) {
        float inv_n = 1.0f / (float)n;
        float mean  = ss[0] * inv_n;
        float var   = sq[0] * inv_n - mean * mean;   // biased, as reference
        stats[0] = mean;
        stats[1] = rsqrtf(var + 1e-5f);
    }
}

// ---------------------------------------------------------------------------
// Kernel 4: BN (train-mode, global stats) + exact-erf GELU + 1x1 conv +
// sigmoid gate; result duplicated into both output halves (att1, att2).
// ---------------------------------------------------------------------------
__global__ void topk_finalize(const float* __restrict__ att,
                              const float* __restrict__ stats,
                              const float* __restrict__ bn_g,
                              const float* __restrict__ bn_b,
                              const float* __restrict__ conv_w,
                              const float* __restrict__ conv_b,
                              float* __restrict__ out, int n) {
    int i = blockIdx.x * blockDim.x + threadIdx.x;
    if (i >= n) return;
    float a = (att[i] - stats[0]) * stats[1] * bn_g[0] + bn_b[0];
    float g = 0.5f * a * (1.0f + erff(a * 0.70710678118654752f));   // exact GELU
    float z = g * conv_w[0] + conv_b[0];
    float gate = 1.0f / (1.0f + expf(-z));
    out[i]     = gate;   // att1
    out[n + i] = gate;   // att2
}

// ---------------------------------------------------------------------------
extern "C" void kernel_launch(void* const* d_in, const int* in_sizes, int n_in,
                              void* d_out, int out_size, void* d_ws, size_t ws_size,
                              hipStream_t stream) {
    const float* x      = (const float*)d_in[0];
    const float* fc_w   = (const float*)d_in[1];
    const float* fc_b   = (const float*)d_in[2];
    const float* bn_g   = (const float*)d_in[3];
    const float* bn_b   = (const float*)d_in[4];
    const float* conv_w = (const float*)d_in[5];
    const float* conv_b = (const float*)d_in[6];

    const int L     = LLEN;
    const int nrows = in_sizes[0] / L;           // B*D = 32768

    // Workspace layout (floats): [stats:2][w:L][att:nrows][psum:64][psumsq:64]
    float* ws     = (float*)d_ws;
    float* stats  = ws;
    float* w      = ws + 2;
    float* att    = ws + 2 + L;
    float* psum   = att + nrows;
    float* psumsq = psum + 64;

    topk_prep_w<<<(L + 255) / 256, 256, 0, stream>>>(fc_w, w, L);

    int gemv_blocks = nrows / 128;               // 8 waves x 16 rows per block
    topk_gemv_wmma<<<gemv_blocks, 256, 0, stream>>>(x, w, fc_b, att, L);

    topk_reduce_partial<<<64, 256, 0, stream>>>(att, psum, psumsq, nrows);
    topk_reduce_final<<<1, 64, 0, stream>>>(psum, psumsq, stats, nrows);

    topk_finalize<<<(nrows + 255) / 256, 256, 0, stream>>>(
        att, stats, bn_g, bn_b, conv_w, conv_b, (float*)d_out, nrows);
}